// HANGAE_28501402977097
// MI455X (gfx1250) — compile-verified
//
#include <hip/hip_runtime.h>
#include <math.h>

// ---------------- problem constants (from reference) ----------------
#define N_NODES 20000
#define N_EDGES 320000
#define M_PATHS 3
#define F_IN    128
#define HEADS1  4
#define C_DIM   64          // C1 == C2 == 64
#define NEG_SLOPE 0.2f
#define BN_EPS  1e-5f

#define ACT_NONE 0
#define ACT_ELU  1
#define ACT_TANH 2

typedef __attribute__((ext_vector_type(2))) float v2f;
typedef __attribute__((ext_vector_type(8))) float v8f;

// ---------------- workspace layout (float offsets) ----------------
static const size_t OFF_STATS = 0;                                        // 64
static const size_t OFF_ECOEF = 64;                                       // 16
static const size_t OFF_HBUF  = 256;                                      // N*256
static const size_t OFF_ACCUM = OFF_HBUF  + (size_t)N_NODES * 256;        // N*256 (also tanh buf N*192)
static const size_t OFF_ALPHA = OFF_ACCUM + (size_t)N_NODES * 256;        // E*4
static const size_t OFF_ASRC  = OFF_ALPHA + (size_t)N_EDGES * 4;          // N*4
static const size_t OFF_ADST  = OFF_ASRC  + (size_t)N_NODES * 4;          // N*4
static const size_t OFF_AMAX  = OFF_ADST  + (size_t)N_NODES * 4;          // N*4 (ordered uint)
static const size_t OFF_DENOM = OFF_AMAX  + (size_t)N_NODES * 4;          // N*4
static const size_t OFF_ZSTK  = OFF_DENOM + (size_t)N_NODES * 4;          // N*3*64
static const size_t OFF_H1F   = OFF_ZSTK  + (size_t)N_NODES * 192;        // N*64
static const size_t OFF_DTMP  = OFF_H1F   + (size_t)N_NODES * 64;         // N*64

// ---------------- small device helpers ----------------
__device__ __forceinline__ unsigned f2ord(float f) {
    unsigned u = __float_as_uint(f);
    return (u & 0x80000000u) ? ~u : (u | 0x80000000u);
}
__device__ __forceinline__ float ord2f(unsigned u) {
    unsigned b = (u & 0x80000000u) ? (u ^ 0x80000000u) : ~u;
    return __uint_as_float(b);
}

// ---------------- templated WMMA fp32 GEMM: C = act(A*B + bias) ----------------
// A [Mr,K] row-major, B [K,NC] row-major. Compile-time K/NC so every fragment
// load is a global_load with an immediate offset off one per-lane base pointer,
// and the K-loop fully unrolls into a straight stream of v_wmma_f32_16x16x4_f32.
// One 16x16 output tile per wave32; tile id is wave-uniform => EXEC stays all-1s.
template <int K, int NC, int ACT>
__global__ void wmma_gemm_f32_kernel(const float* __restrict__ A,
                                     const float* __restrict__ B,
                                     const float* __restrict__ bias,
                                     float* __restrict__ C, int Mr)
{
    constexpr int ntn = NC >> 4;
    const int ntiles  = (Mr >> 4) * ntn;
    const int tile    = blockIdx.x * (blockDim.x >> 5) + (threadIdx.x >> 5);
    if (tile >= ntiles) return;            // uniform across the wave

    const int lane = threadIdx.x & 31;
    const int half = lane >> 4;            // lanes 16..31 hold the K+2 / M+8 halves
    const int l    = lane & 15;
    const int tm   = (tile / ntn) << 4;
    const int tn   = (tile % ntn) << 4;

    v8f acc = {};
    const float* __restrict__ arow = A + (size_t)(tm + l) * K + (half << 1);
    const float* __restrict__ bcol = B + (size_t)(half << 1) * NC + tn + l;
#pragma unroll
    for (int k0 = 0; k0 < K; k0 += 4) {
        v2f a; a.x = arow[k0];           a.y = arow[k0 + 1];
        v2f b; b.x = bcol[k0 * NC];      b.y = bcol[k0 * NC + NC];
        acc = __builtin_amdgcn_wmma_f32_16x16x4_f32(
                  false, a, false, b, (short)0, acc, false, false);
    }
    const int   col = tn + l;
    const float bv  = bias ? bias[col] : 0.0f;
    float* __restrict__ crow = C + (size_t)(tm + (half << 3)) * NC + col;
#pragma unroll
    for (int r = 0; r < 8; ++r) {
        float v = acc[r] + bv;
        if (ACT == ACT_ELU)  v = (v > 0.0f) ? v : expm1f(v);
        if (ACT == ACT_TANH) v = tanhf(v);
        crow[(size_t)r * NC] = v;
    }
}

// ---------------- utility: zero floats ----------------
__global__ void zero_f32_kernel(float* p, size_t n) {
    for (size_t i = blockIdx.x * (size_t)blockDim.x + threadIdx.x; i < n;
         i += (size_t)gridDim.x * blockDim.x)
        p[i] = 0.0f;
}

// ---------------- BN stats over log1p(edge_attr[m]) ----------------
__global__ void bn_stats_kernel(const float* __restrict__ attr, float* __restrict__ stats) {
    const int m = blockIdx.y;
    const float* a = attr + (size_t)m * N_EDGES;
    float s = 0.0f, sq = 0.0f;
    for (int i = blockIdx.x * blockDim.x + threadIdx.x; i < N_EDGES;
         i += gridDim.x * blockDim.x) {
        float v = log1pf(a[i]);
        s += v; sq += v * v;
    }
    __shared__ float ls[256], lq[256];
    ls[threadIdx.x] = s; lq[threadIdx.x] = sq;
    __syncthreads();
    for (int st = 128; st > 0; st >>= 1) {
        if (threadIdx.x < st) {
            ls[threadIdx.x] += ls[threadIdx.x + st];
            lq[threadIdx.x] += lq[threadIdx.x + st];
        }
        __syncthreads();
    }
    if (threadIdx.x == 0) {
        atomicAdd(&stats[m * 16 + 0], ls[0]);
        atomicAdd(&stats[m * 16 + 1], lq[0]);
    }
}

__global__ void bn_finalize_kernel(float* stats) {
    const int m = threadIdx.x;
    if (m < M_PATHS) {
        float mean = stats[m * 16 + 0] / (float)N_EDGES;
        float var  = stats[m * 16 + 1] / (float)N_EDGES - mean * mean;
        stats[m * 16 + 2] = mean;
        stats[m * 16 + 3] = rsqrtf(var + BN_EPS);
    }
}

// ---------------- per-head edge coefficient: dot(lin_edge_w[h], att_edge[h]) ----------------
__global__ void edge_coef_kernel(const float* __restrict__ lin_edge_w,
                                 const float* __restrict__ att_edge,
                                 float* __restrict__ ecoef, int H) {
    const int h = threadIdx.x;
    if (h < H) {
        float s = 0.0f;
        for (int c = 0; c < C_DIM; ++c)
            s += lin_edge_w[h * C_DIM + c] * att_edge[h * C_DIM + c];
        ecoef[h] = s;
    }
}

// ---------------- per-node attention projections ----------------
template <int H>
__global__ void node_att_kernel(const float* __restrict__ hbuf,
                                const float* __restrict__ att_src,
                                const float* __restrict__ att_dst,
                                float* __restrict__ asrc, float* __restrict__ adst) {
    const int idx = blockIdx.x * blockDim.x + threadIdx.x;   // n*H + h
    if (idx >= N_NODES * H) return;
    const int h = idx & (H - 1);
    const float* hv = hbuf + (size_t)idx * C_DIM;
    float s = 0.0f, d = 0.0f;
#pragma unroll 8
    for (int c = 0; c < C_DIM; ++c) {
        float v = hv[c];
        s += v * att_src[h * C_DIM + c];
        d += v * att_dst[h * C_DIM + c];
    }
    asrc[idx] = s; adst[idx] = d;
}

// ---------------- edge alpha: leaky_relu + segment-max (ordered-uint atomicMax) ----------------
template <int H>
__global__ void alpha_kernel(const int* __restrict__ src, const int* __restrict__ dst,
                             const float* __restrict__ attr,
                             const float* __restrict__ stats_m,
                             const float* __restrict__ gamma, const float* __restrict__ beta_,
                             const float* __restrict__ ecoef,
                             const float* __restrict__ asrc, const float* __restrict__ adst,
                             float* __restrict__ alpha, unsigned* __restrict__ amax) {
    const float mean = stats_m[2], rstd = stats_m[3];
    const float g = gamma[0], b = beta_[0];
    float ec[H];
#pragma unroll
    for (int h = 0; h < H; ++h) ec[h] = ecoef[h];
    const int stride = gridDim.x * blockDim.x;
    for (int e = blockIdx.x * blockDim.x + threadIdx.x; e < N_EDGES; e += stride) {
        if (e + stride < N_EDGES) {                // hide gather latency
            __builtin_prefetch(&src[e + stride], 0, 0);
            __builtin_prefetch(&dst[e + stride], 0, 0);
        }
        const int s = src[e], d = dst[e];
        const float ea = (log1pf(attr[e]) - mean) * rstd * g + b;
#pragma unroll
        for (int h = 0; h < H; ++h) {
            float al = asrc[s * H + h] + adst[d * H + h] + ea * ec[h];
            al = (al > 0.0f) ? al : NEG_SLOPE * al;
            alpha[(size_t)e * H + h] = al;
            atomicMax(&amax[d * H + h], f2ord(al));
        }
    }
}

// ---------------- exp(alpha - max) and segment-sum denominator ----------------
template <int H>
__global__ void exp_denom_kernel(const int* __restrict__ dst, float* __restrict__ alpha,
                                 const unsigned* __restrict__ amax,
                                 float* __restrict__ denom) {
    for (int e = blockIdx.x * blockDim.x + threadIdx.x; e < N_EDGES;
         e += gridDim.x * blockDim.x) {
        const int d = dst[e];
#pragma unroll
        for (int h = 0; h < H; ++h) {
            const float am = ord2f(amax[d * H + h]);
            const float ex = expf(alpha[(size_t)e * H + h] - am);
            alpha[(size_t)e * H + h] = ex;
            atomicAdd(&denom[d * H + h], ex);
        }
    }
}

// ---------------- normalize edge weights ----------------
template <int H>
__global__ void norm_alpha_kernel(const int* __restrict__ dst, float* __restrict__ alpha,
                                  const float* __restrict__ denom) {
    for (int e = blockIdx.x * blockDim.x + threadIdx.x; e < N_EDGES;
         e += gridDim.x * blockDim.x) {
        const int d = dst[e];
#pragma unroll
        for (int h = 0; h < H; ++h)
            alpha[(size_t)e * H + h] /= (denom[d * H + h] + 1e-16f);
    }
}

// ---------------- weighted scatter-add of h[src] into accum[dst] ----------------
template <int H>
__global__ void scatter_kernel(const int* __restrict__ src, const int* __restrict__ dst,
                               const float* __restrict__ alpha,
                               const float* __restrict__ hbuf,
                               float* __restrict__ accum) {
    const size_t total = (size_t)N_EDGES * C_DIM;
    for (size_t i = blockIdx.x * (size_t)blockDim.x + threadIdx.x; i < total;
         i += (size_t)gridDim.x * blockDim.x) {
        const int e = (int)(i >> 6), c = (int)(i & 63);
        const int s = src[e], d = dst[e];
#pragma unroll
        for (int h = 0; h < H; ++h) {
            const float w = alpha[(size_t)e * H + h];
            atomicAdd(&accum[((size_t)d * H + h) * C_DIM + c],
                      hbuf[((size_t)s * H + h) * C_DIM + c] * w);
        }
    }
}

// ---------------- head-mean + bias + ELU into z_stack slot m ----------------
template <int H>
__global__ void finalize_gat_kernel(const float* __restrict__ accum,
                                    const float* __restrict__ bias,
                                    float* __restrict__ zstk, int m) {
    const int i = blockIdx.x * blockDim.x + threadIdx.x;   // n*64 + c
    if (i >= N_NODES * C_DIM) return;
    const int n = i >> 6, c = i & 63;
    float s = 0.0f;
#pragma unroll
    for (int h = 0; h < H; ++h) s += accum[((size_t)n * H + h) * C_DIM + c];
    s = s * (1.0f / H) + bias[c];
    s = (s > 0.0f) ? s : expm1f(s);        // elu
    zstk[((size_t)n * M_PATHS + m) * C_DIM + c] = s;
}

// ---------------- semantic attention reduce (softmax over M, weighted sum) ----------------
__global__ void sem_reduce_kernel(const float* __restrict__ tanh_buf,
                                  const float* __restrict__ sem_vec,
                                  const float* __restrict__ zstk,
                                  float* __restrict__ zout, float* __restrict__ beta_out) {
    const int n = blockIdx.x * blockDim.x + threadIdx.x;
    if (n >= N_NODES) return;
    float sc[M_PATHS];
    float mx = -1e30f;
    for (int m = 0; m < M_PATHS; ++m) {
        const float* t = tanh_buf + ((size_t)n * M_PATHS + m) * C_DIM;
        float s = 0.0f;
        for (int c = 0; c < C_DIM; ++c) s += t[c] * sem_vec[c];
        sc[m] = s; mx = fmaxf(mx, s);
    }
    float den = 0.0f;
    for (int m = 0; m < M_PATHS; ++m) { sc[m] = expf(sc[m] - mx); den += sc[m]; }
    for (int m = 0; m < M_PATHS; ++m) sc[m] /= den;
    for (int c = 0; c < C_DIM; ++c) {
        float v = 0.0f;
        for (int m = 0; m < M_PATHS; ++m)
            v += zstk[((size_t)n * M_PATHS + m) * C_DIM + c] * sc[m];
        zout[(size_t)n * C_DIM + c] = v;
    }
    if (beta_out)
        for (int m = 0; m < M_PATHS; ++m) beta_out[n * M_PATHS + m] = sc[m];
}

// ---------------- host-side helpers ----------------
static inline void zero_f32(hipStream_t st, float* p, size_t n) {
    unsigned gb = (unsigned)((n + 255) / 256);
    if (gb > 8192u) gb = 8192u;
    zero_f32_kernel<<<gb, 256, 0, st>>>(p, n);
}

template <int K, int NC>
static void gemm(hipStream_t st, const float* A, const float* B, const float* bias,
                 float* C, int Mr, int act) {
    const int ntiles = (Mr >> 4) * (NC >> 4);
    const int blocks = (ntiles + 7) / 8;            // 8 waves (tiles) per 256-thread block
    switch (act) {
    case ACT_ELU:
        wmma_gemm_f32_kernel<K, NC, ACT_ELU><<<blocks, 256, 0, st>>>(A, B, bias, C, Mr); break;
    case ACT_TANH:
        wmma_gemm_f32_kernel<K, NC, ACT_TANH><<<blocks, 256, 0, st>>>(A, B, bias, C, Mr); break;
    default:
        wmma_gemm_f32_kernel<K, NC, ACT_NONE><<<blocks, 256, 0, st>>>(A, B, bias, C, Mr); break;
    }
}

template <int FIN, int H>
static void run_han_layer(hipStream_t st, float* ws,
                          const float* xin,
                          const int* ei_all, const float* eattr_all,
                          const float* lin_w, size_t lin_stride,
                          const float* att_src, const float* att_dst, size_t att_stride,
                          const float* lin_edge, size_t le_stride,
                          const float* att_edge, size_t ae_stride,
                          const float* bias, size_t bias_stride,
                          const float* bng, const float* bnb,
                          const float* sem_w, const float* sem_b, const float* sem_vec,
                          float* zout, float* beta_out) {
    float* stats = ws + OFF_STATS;
    float* ecoef = ws + OFF_ECOEF;
    float* hbuf  = ws + OFF_HBUF;
    float* accum = ws + OFF_ACCUM;
    float* alpha = ws + OFF_ALPHA;
    float* asrc  = ws + OFF_ASRC;
    float* adst  = ws + OFF_ADST;
    unsigned* amax = (unsigned*)(ws + OFF_AMAX);
    float* denom = ws + OFF_DENOM;
    float* zstk  = ws + OFF_ZSTK;

    for (int m = 0; m < M_PATHS; ++m) {
        const int* src = ei_all + (size_t)m * 2 * N_EDGES;
        const int* dst = src + N_EDGES;
        const float* attr = eattr_all + (size_t)m * N_EDGES;

        edge_coef_kernel<<<1, 32, 0, st>>>(lin_edge + m * le_stride,
                                           att_edge + m * ae_stride, ecoef, H);
        // h = xin @ lin_w[m]   (fp32 WMMA, fully unrolled K-loop)
        gemm<FIN, H * C_DIM>(st, xin, lin_w + m * lin_stride, nullptr, hbuf,
                             N_NODES, ACT_NONE);

        node_att_kernel<H><<<(N_NODES * H + 255) / 256, 256, 0, st>>>(
            hbuf, att_src + m * att_stride, att_dst + m * att_stride, asrc, adst);

        zero_f32(st, (float*)amax, (size_t)N_NODES * H);   // ordered-uint 0 == "-inf"
        zero_f32(st, denom, (size_t)N_NODES * H);
        zero_f32(st, accum, (size_t)N_NODES * H * C_DIM);

        alpha_kernel<H><<<1024, 256, 0, st>>>(src, dst, attr, stats + m * 16,
                                              bng + m, bnb + m, ecoef, asrc, adst,
                                              alpha, amax);
        exp_denom_kernel<H><<<1024, 256, 0, st>>>(dst, alpha, amax, denom);
        norm_alpha_kernel<H><<<1024, 256, 0, st>>>(dst, alpha, denom);
        scatter_kernel<H><<<8192, 256, 0, st>>>(src, dst, alpha, hbuf, accum);
        finalize_gat_kernel<H><<<(N_NODES * C_DIM + 255) / 256, 256, 0, st>>>(
            accum, bias + m * bias_stride, zstk, m);
    }
    // semantic attention: tanh(z_stack @ sem_w + sem_b) into accum (reused as [N*3,64])
    gemm<C_DIM, C_DIM>(st, zstk, sem_w, sem_b, accum, N_NODES * M_PATHS, ACT_TANH);
    sem_reduce_kernel<<<(N_NODES + 255) / 256, 256, 0, st>>>(accum, sem_vec, zstk,
                                                             zout, beta_out);
}

// ---------------- entry point ----------------
extern "C" void kernel_launch(void* const* d_in, const int* in_sizes, int n_in,
                              void* d_out, int out_size, void* d_ws, size_t ws_size,
                              hipStream_t stream) {
    (void)in_sizes; (void)n_in; (void)out_size; (void)ws_size;
    const float* x          = (const float*)d_in[0];
    const int*   edge_index = (const int*)  d_in[1];
    const float* edge_attr  = (const float*)d_in[2];
    const float* lin1_w     = (const float*)d_in[3];
    const float* att1_src   = (const float*)d_in[4];
    const float* att1_dst   = (const float*)d_in[5];
    const float* lin1_edge  = (const float*)d_in[6];
    const float* att1_edge  = (const float*)d_in[7];
    const float* bias1      = (const float*)d_in[8];
    const float* bn1_g      = (const float*)d_in[9];
    const float* bn1_b      = (const float*)d_in[10];
    const float* sem1_w     = (const float*)d_in[11];
    const float* sem1_b     = (const float*)d_in[12];
    const float* sem1_vec   = (const float*)d_in[13];
    const float* lin2_w     = (const float*)d_in[14];
    const float* att2_src   = (const float*)d_in[15];
    const float* att2_dst   = (const float*)d_in[16];
    const float* lin2_edge  = (const float*)d_in[17];
    const float* att2_edge  = (const float*)d_in[18];
    const float* bias2      = (const float*)d_in[19];
    const float* bn2_g      = (const float*)d_in[20];
    const float* bn2_b      = (const float*)d_in[21];
    const float* sem2_w     = (const float*)d_in[22];
    const float* sem2_b     = (const float*)d_in[23];
    const float* sem2_vec   = (const float*)d_in[24];
    const float* dec1_w     = (const float*)d_in[25];
    const float* dec1_b     = (const float*)d_in[26];
    const float* dec2_w     = (const float*)d_in[27];
    const float* dec2_b     = (const float*)d_in[28];

    float* ws   = (float*)d_ws;
    float* outf = (float*)d_out;
    float* z_out    = outf;                                  // [N,64]
    float* xhat_out = outf + (size_t)N_NODES * C_DIM;        // [N,128]
    float* beta_out = xhat_out + (size_t)N_NODES * F_IN;     // [N,3]

    // BN statistics of log1p(edge_attr[m]) — shared by both layers.
    zero_f32(stream, ws + OFF_STATS, 64);
    bn_stats_kernel<<<dim3(256, M_PATHS), 256, 0, stream>>>(edge_attr, ws + OFF_STATS);
    bn_finalize_kernel<<<1, 32, 0, stream>>>(ws + OFF_STATS);

    // ----- HAN layer 1 (heads = 4) -> h1_final [N,64] -----
    run_han_layer<F_IN, HEADS1>(stream, ws, x, edge_index, edge_attr,
                                lin1_w, (size_t)F_IN * HEADS1 * C_DIM,
                                att1_src, att1_dst, (size_t)HEADS1 * C_DIM,
                                lin1_edge, (size_t)HEADS1 * C_DIM,
                                att1_edge, (size_t)HEADS1 * C_DIM,
                                bias1, (size_t)C_DIM,
                                bn1_g, bn1_b, sem1_w, sem1_b, sem1_vec,
                                ws + OFF_H1F, nullptr);

    // ----- HAN layer 2 (heads = 1) -> z [N,64] + beta_w [N,3] (direct to d_out) -----
    run_han_layer<C_DIM, 1>(stream, ws, ws + OFF_H1F, edge_index, edge_attr,
                            lin2_w, (size_t)C_DIM * C_DIM,
                            att2_src, att2_dst, (size_t)C_DIM,
                            lin2_edge, (size_t)C_DIM,
                            att2_edge, (size_t)C_DIM,
                            bias2, (size_t)C_DIM,
                            bn2_g, bn2_b, sem2_w, sem2_b, sem2_vec,
                            z_out, beta_out);

    // ----- decoder MLP: x_hat = elu(z @ dec1 + b1) @ dec2 + b2 -----
    gemm<C_DIM, C_DIM>(stream, z_out, dec1_w, dec1_b, ws + OFF_DTMP, N_NODES, ACT_ELU);
    gemm<C_DIM, F_IN>(stream, ws + OFF_DTMP, dec2_w, dec2_b, xhat_out, N_NODES, ACT_NONE);
}